// SparsePheromonAttention_34875134443698
// MI455X (gfx1250) — compile-verified
//
#include <hip/hip_runtime.h>
#include <math.h>

#define GAMMA_C   2.5f
#define RHO_C     0.085f
#define TAUCLIP_C 5.0f

typedef __attribute__((ext_vector_type(2))) float v2f;
typedef __attribute__((ext_vector_type(8))) float v8f;

// ---------------------------------------------------------------------------
// GEMM: C[M,N] = A[M,K] @ B[K,N] + bias[N], all fp32, via V_WMMA_F32_16X16X4_F32.
// One wave32 computes a 16x64 output strip: 4 accumulators sharing one A
// fragment per K-step (A traffic /4, 4 independent WMMAs in flight).
// M multiple of 16, N multiple of 64, K multiple of 4.
// A-frag (ISA 7.12.2, 32-bit A 16x4): lanes 0-15 hold K={k0,k0+1} of row M=lane,
// lanes 16-31 hold K={k0+2,k0+3}. B-frag mirrored. C/D: VGPR r -> M=r
// (lanes 0-15) / M=r+8 (lanes 16-31), N=lane&15.
// ---------------------------------------------------------------------------
__global__ void wmma_gemm_bias_f32(const float* __restrict__ A,
                                   const float* __restrict__ B,
                                   const float* __restrict__ bias,
                                   float* __restrict__ C,
                                   int M, int N, int K)
{
    const int lane = threadIdx.x & 31;
    const int wave = (int)((blockIdx.x * blockDim.x + threadIdx.x) >> 5);
    const int ntn  = N >> 6;                  // 64-wide N strips
    const int mt   = wave / ntn;
    const int nt   = wave - mt * ntn;
    if (mt * 16 >= M) return;                 // uniform per wave (EXEC stays all-1s)

    const int half = lane >> 4;               // 0 -> K pair (0,1), 1 -> (2,3)
    const int l15  = lane & 15;

    const float* Arow  = A + (size_t)(mt * 16 + l15) * K + 2 * half;
    const float* Bbase = B + (size_t)(2 * half) * N + (nt * 64 + l15);

    v8f acc0 = {0.f,0.f,0.f,0.f,0.f,0.f,0.f,0.f};
    v8f acc1 = acc0, acc2 = acc0, acc3 = acc0;

    for (int k0 = 0; k0 < K; k0 += 4) {
        v2f a;
        a.x = Arow[k0];
        a.y = Arow[k0 + 1];
        const float* Bk = Bbase + (size_t)k0 * N;
        v2f b0, b1, b2, b3;
        b0.x = Bk[0];       b0.y = Bk[N];
        b1.x = Bk[16];      b1.y = Bk[N + 16];
        b2.x = Bk[32];      b2.y = Bk[N + 32];
        b3.x = Bk[48];      b3.y = Bk[N + 48];
        acc0 = __builtin_amdgcn_wmma_f32_16x16x4_f32(false, a, false, b0,
                                                     (short)0, acc0, false, false);
        acc1 = __builtin_amdgcn_wmma_f32_16x16x4_f32(false, a, false, b1,
                                                     (short)0, acc1, false, false);
        acc2 = __builtin_amdgcn_wmma_f32_16x16x4_f32(false, a, false, b2,
                                                     (short)0, acc2, false, false);
        acc3 = __builtin_amdgcn_wmma_f32_16x16x4_f32(false, a, false, b3,
                                                     (short)0, acc3, false, false);
    }

    float* Crow = C + (size_t)(mt * 16 + 8 * half) * N + (nt * 64 + l15);
#pragma unroll
    for (int t = 0; t < 4; ++t) {
        const v8f acc = (t == 0) ? acc0 : (t == 1) ? acc1 : (t == 2) ? acc2 : acc3;
        const int   n  = nt * 64 + 16 * t + l15;
        const float bv = bias ? bias[n] : 0.f;
#pragma unroll
        for (int r = 0; r < 8; ++r)
            Crow[(size_t)r * N + 16 * t] = acc[r] + bv;
    }
}

// ---------------------------------------------------------------------------
// Top-4 of each tau row (jax top_k tie-break: lower index wins on equal value)
// plus the row sum. One wave32 per row, with stream prefetch.
// ---------------------------------------------------------------------------
__device__ __forceinline__ void top4_insert(float v, int i, float* val, int* id)
{
    if (v > val[3] || (v == val[3] && i < id[3])) {
        val[3] = v; id[3] = i;
#pragma unroll
        for (int j = 3; j > 0; --j) {
            bool sw = (val[j] > val[j-1]) || (val[j] == val[j-1] && id[j] < id[j-1]);
            if (sw) {
                float tv = val[j]; val[j] = val[j-1]; val[j-1] = tv;
                int   ti = id[j];  id[j]  = id[j-1];  id[j-1]  = ti;
            }
        }
    }
}

__global__ void topk_rowsum_kernel(const float* __restrict__ tau,
                                   int T, int nrows,
                                   int* __restrict__ idx_out,
                                   float* __restrict__ rowsum_out)
{
    const int lane = threadIdx.x & 31;
    const int row  = (int)((blockIdx.x * blockDim.x + threadIdx.x) >> 5);
    if (row >= nrows) return;

    const float* trow = tau + (size_t)row * T;
    float val[4] = {-INFINITY, -INFINITY, -INFINITY, -INFINITY};
    int   id[4]  = {0x7fffffff, 0x7fffffff, 0x7fffffff, 0x7fffffff};
    float sum = 0.f;

    for (int s = lane; s < T; s += 32) {
        if (s + 256 < T) __builtin_prefetch(&trow[s + 256], 0, 1);  // global_prefetch_b8
        float v = trow[s];
        sum += v;
        top4_insert(v, s, val, id);
    }
#pragma unroll
    for (int off = 16; off > 0; off >>= 1) {
        sum += __shfl_xor(sum, off);
        float ov[4]; int oi[4];
#pragma unroll
        for (int j = 0; j < 4; ++j) {
            ov[j] = __shfl_xor(val[j], off);
            oi[j] = __shfl_xor(id[j],  off);
        }
#pragma unroll
        for (int j = 0; j < 4; ++j) top4_insert(ov[j], oi[j], val, id);
    }
    if (lane == 0) {
#pragma unroll
        for (int j = 0; j < 4; ++j) idx_out[row * 4 + j] = id[j];
        rowsum_out[row] = sum;
    }
}

// ---------------------------------------------------------------------------
// Sparse attention: one wave32 per (b,t) row. For each of H=8 heads compute
// 4 logits (64-dot via shuffle reduce), softmax over 4, accumulate out_heads,
// and the head-mean signal^3 per selected key. D must be 64.
// ---------------------------------------------------------------------------
__global__ void sparse_attn_kernel(const float* __restrict__ qkv,
                                   const float* __restrict__ tau,
                                   const int* __restrict__ topidx,
                                   float* __restrict__ out_heads,
                                   float* __restrict__ sig3,
                                   int T, int C, int nrows)
{
    const int lane = threadIdx.x & 31;
    const int row  = (int)((blockIdx.x * blockDim.x + threadIdx.x) >> 5);
    if (row >= nrows) return;

    const int    D       = 64;
    const int    H       = C / D;            // 8
    const size_t stride  = (size_t)3 * C;    // qkv row stride
    const int    rowbase = (row / T) * T;    // b*T
    const float  scale   = 0.125f;           // 64^-0.5
    const float* qrow    = qkv + (size_t)row * stride;

    int id[4]; float bias[4];
#pragma unroll
    for (int j = 0; j < 4; ++j) {
        id[j]   = topidx[row * 4 + j];
        bias[j] = GAMMA_C * logf(tau[(size_t)row * T + id[j]] + 1e-8f);
    }

    float sig[4] = {0.f, 0.f, 0.f, 0.f};
    for (int h = 0; h < H; ++h) {
        const float* qh = qrow + h * D;
        const float  q0 = qh[lane], q1 = qh[lane + 32];

        float logit[4];
#pragma unroll
        for (int j = 0; j < 4; ++j) {
            const float* kh = qkv + (size_t)(rowbase + id[j]) * stride + C + h * D;
            float p = q0 * kh[lane] + q1 * kh[lane + 32];
#pragma unroll
            for (int off = 16; off > 0; off >>= 1) p += __shfl_xor(p, off);
            logit[j] = p * scale + bias[j];
        }

        float mx = fmaxf(fmaxf(logit[0], logit[1]), fmaxf(logit[2], logit[3]));
        float e[4], se = 0.f;
#pragma unroll
        for (int j = 0; j < 4; ++j) { e[j] = expf(logit[j] - mx); se += e[j]; }
        const float inv = 1.f / se;

        float o0 = 0.f, o1 = 0.f;
#pragma unroll
        for (int j = 0; j < 4; ++j) {
            const float* vh = qkv + (size_t)(rowbase + id[j]) * stride + 2 * C + h * D;
            const float pj = e[j] * inv;
            o0 += pj * vh[lane];
            o1 += pj * vh[lane + 32];
            sig[j] += pj;
        }
        out_heads[(size_t)row * C + h * D + lane]      = o0;
        out_heads[(size_t)row * C + h * D + lane + 32] = o1;
    }

    if (lane == 0) {
        const float invH = 1.f / (float)H;
#pragma unroll
        for (int j = 0; j < 4; ++j) {
            float s = sig[j] * invH;       // mean over heads
            sig3[row * 4 + j] = s * s * s; // signal^3
        }
    }
}

// ---------------------------------------------------------------------------
// tau_new = min( ((1-rho)*tau + sig3 at topk) / (rowsum' + 1e-8), TAU_CLIP )
// One 256-thread block per row, with stream prefetch.
// ---------------------------------------------------------------------------
__global__ void tau_update_kernel(const float* __restrict__ tau,
                                  const int* __restrict__ topidx,
                                  const float* __restrict__ sig3,
                                  const float* __restrict__ rowsum,
                                  float* __restrict__ tau_new,
                                  int T)
{
    const int row = blockIdx.x;
    const int i0 = topidx[row*4+0], i1 = topidx[row*4+1];
    const int i2 = topidx[row*4+2], i3 = topidx[row*4+3];
    const float s0 = sig3[row*4+0], s1 = sig3[row*4+1];
    const float s2 = sig3[row*4+2], s3 = sig3[row*4+3];

    const float keep  = 1.f - RHO_C;
    const float denom = keep * rowsum[row] + s0 + s1 + s2 + s3 + 1e-8f;
    const float inv   = 1.f / denom;

    const float* trow = tau     + (size_t)row * T;
    float*       nrow = tau_new + (size_t)row * T;
    for (int s = threadIdx.x; s < T; s += blockDim.x) {
        if (s + 2 * (int)blockDim.x < T)
            __builtin_prefetch(&trow[s + 2 * blockDim.x], 0, 1);
        float v = keep * trow[s];
        v += (s == i0) ? s0 : 0.f;
        v += (s == i1) ? s1 : 0.f;
        v += (s == i2) ? s2 : 0.f;
        v += (s == i3) ? s3 : 0.f;
        nrow[s] = fminf(v * inv, TAUCLIP_C);
    }
}

// ---------------------------------------------------------------------------
extern "C" void kernel_launch(void* const* d_in, const int* in_sizes, int n_in,
                              void* d_out, int out_size, void* d_ws, size_t ws_size,
                              hipStream_t stream)
{
    const float* x     = (const float*)d_in[0];
    const float* tau   = (const float*)d_in[1];
    const float* Wqkv  = (const float*)d_in[2];
    const float* bqkv  = (const float*)d_in[3];
    const float* Wproj = (const float*)d_in[4];
    const float* bproj = (const float*)d_in[5];

    // Derive dims: Wqkv has 3*C*C elements; x has B*T*C; tau has B*T*T.
    const long n_x   = in_sizes[0];
    const long n_tau = in_sizes[1];
    const long n_wq  = in_sizes[2];
    const int  C     = (int)lround(sqrt((double)n_wq / 3.0));  // 512
    const long BT    = n_x / C;                                 // B*T = 4096
    const int  T     = (int)(n_tau / BT);                       // 2048
    const int  nrows = (int)BT;

    // Workspace layout
    float* qkv       = (float*)d_ws;                             // BT x 3C
    float* out_heads = qkv + (size_t)BT * 3 * C;                 // BT x C
    int*   topidx    = (int*)(out_heads + (size_t)BT * C);       // BT x 4
    float* rowsum    = (float*)(topidx + (size_t)BT * 4);        // BT
    float* sig3      = rowsum + BT;                              // BT x 4

    float* out_final = (float*)d_out;                 // [B,T,C]
    float* tau_new   = out_final + (size_t)BT * C;    // [B,T,T]

    // 1) qkv = x @ Wqkv + bqkv
    {
        const int  M = nrows, N = 3 * C, K = C;
        const long waves  = (long)(M / 16) * (N / 64);
        const long blocks = (waves + 7) / 8;
        wmma_gemm_bias_f32<<<dim3((unsigned)blocks), dim3(256), 0, stream>>>(
            x, Wqkv, bqkv, qkv, M, N, K);
    }
    // 2) top-4 indices + row sums of tau
    {
        const long blocks = (nrows + 7) / 8;
        topk_rowsum_kernel<<<dim3((unsigned)blocks), dim3(256), 0, stream>>>(
            tau, T, nrows, topidx, rowsum);
    }
    // 3) sparse attention + signal^3
    {
        const long blocks = (nrows + 7) / 8;
        sparse_attn_kernel<<<dim3((unsigned)blocks), dim3(256), 0, stream>>>(
            qkv, tau, topidx, out_heads, sig3, T, C, nrows);
    }
    // 4) out = out_heads @ Wproj + bproj
    {
        const int  M = nrows, N = C, K = C;
        const long waves  = (long)(M / 16) * (N / 64);
        const long blocks = (waves + 7) / 8;
        wmma_gemm_bias_f32<<<dim3((unsigned)blocks), dim3(256), 0, stream>>>(
            out_heads, Wproj, bproj, out_final, M, N, K);
    }
    // 5) pheromone EMA update, normalize, clip
    tau_update_kernel<<<dim3((unsigned)nrows), dim3(256), 0, stream>>>(
        tau, topidx, sig3, rowsum, tau_new, T);
}